// SparseAddmmOp_73710228734302
// MI455X (gfx1250) — compile-verified
//
#include <hip/hip_runtime.h>
#include <stdint.h>

#define M_DIM 16384
#define K_DIM 16384
#define N_DIM 256
#define NNZ_CNT (1u << 20)
#define RPG 8  // rows gathered per TDM op

typedef uint32_t u32;
typedef uint32_t v4u __attribute__((ext_vector_type(4)));
typedef int32_t  v4i __attribute__((ext_vector_type(4)));
typedef int32_t  v8i __attribute__((ext_vector_type(8)));

#ifndef __has_builtin
#define __has_builtin(x) 0
#endif

#if __has_builtin(__builtin_amdgcn_tensor_load_to_lds) && __has_builtin(__builtin_amdgcn_s_wait_tensorcnt)
#define HAVE_TDM 1
#endif

// ---------------- binning kernels ----------------

__global__ void zero_counts(int* counts) {
    int i = blockIdx.x * blockDim.x + threadIdx.x;
    if (i < M_DIM) counts[i] = 0;
}

__global__ void count_rows(const int* __restrict__ rows, int* __restrict__ counts) {
    int i = blockIdx.x * blockDim.x + threadIdx.x;
    if (i < (int)NNZ_CNT) atomicAdd(&counts[rows[i]], 1);
}

// single block of 1024 threads, 16 counts per thread, Hillis-Steele over partials
__global__ void scan_counts(const int* __restrict__ counts, int* __restrict__ seg,
                            int* __restrict__ cursor) {
    __shared__ int partial[1024];
    int t = threadIdx.x;
    int base = t * 16;
    int loc[16];
    int s = 0;
#pragma unroll
    for (int i = 0; i < 16; ++i) { loc[i] = counts[base + i]; s += loc[i]; }
    partial[t] = s;
    __syncthreads();
    for (int off = 1; off < 1024; off <<= 1) {
        int v = partial[t];
        int u = (t >= off) ? partial[t - off] : 0;
        __syncthreads();
        partial[t] = v + u;
        __syncthreads();
    }
    int ex = (t == 0) ? 0 : partial[t - 1];
#pragma unroll
    for (int i = 0; i < 16; ++i) {
        seg[base + i] = ex;
        cursor[base + i] = ex;
        ex += loc[i];
    }
    if (t == 1023) seg[M_DIM] = ex;
}

__global__ void scatter_nnz(const float* __restrict__ values, const int* __restrict__ rows,
                            const int* __restrict__ cols, int* __restrict__ cursor,
                            float* __restrict__ sval, int* __restrict__ scol) {
    int i = blockIdx.x * blockDim.x + threadIdx.x;
    if (i < (int)NNZ_CNT) {
        int r = rows[i];
        int p = atomicAdd(&cursor[r], 1);
        sval[p] = values[i];
        scol[p] = cols[i];
    }
}

// ---------------- accumulation: one wave32 per output row ----------------

#ifdef HAVE_TDM
__device__ __forceinline__ u32 lds_off_u32(const void* p) {
    return (u32)(uintptr_t)(const __attribute__((address_space(3))) void*)p;
}

// TDM gather-mode: each TENSOR_LOAD_TO_LDS pulls RPG gathered rows of `dense`
// (8 KB) into an LDS double buffer; s_wait_tensorcnt synchronizes. EXEC is
// ignored by TDM and TENSORcnt is per-wave, so each wave drives its own DMA.
__global__ void __launch_bounds__(256) spmm_rows_tdm(
    const float* __restrict__ input, const float* __restrict__ dense,
    const int* __restrict__ seg, const float* __restrict__ sval,
    const int* __restrict__ scol, float* __restrict__ out) {
    extern __shared__ float smem[];  // 8 waves * 2 bufs * RPG rows * 256 f32 = 128 KB
    const int lane = threadIdx.x & 31;
    const int wave = threadIdx.x >> 5;
    const int row  = blockIdx.x * 8 + wave;
    const int s = seg[row], e = seg[row + 1];
    const int len = e - s;
    const int nch = (len + RPG - 1) / RPG;

    float* buf0 = &smem[(wave * 2 + 0) * RPG * N_DIM];
    float* buf1 = &smem[(wave * 2 + 1) * RPG * N_DIM];
    const u32 lds0 = __builtin_amdgcn_readfirstlane((int)lds_off_u32(buf0));
    const u32 lds1 = __builtin_amdgcn_readfirstlane((int)lds_off_u32(buf1));
    const uint64_t gbase = (uint64_t)(uintptr_t)dense;

    float4 acc0 = make_float4(0.f, 0.f, 0.f, 0.f);
    float4 acc1 = make_float4(0.f, 0.f, 0.f, 0.f);

    auto issue = [&](int ch) {
        // 8 column indices for this chunk (clamped; repeats are legal in gather mode)
        int jj = s + ch * RPG + (lane & 7);
        if (jj > e - 1) jj = e - 1;
        int c = scol[jj];
        u32 i0 = ((u32)__builtin_amdgcn_readlane(c, 0) & 0xFFFFu) |
                 ((u32)__builtin_amdgcn_readlane(c, 1) << 16);
        u32 i1 = ((u32)__builtin_amdgcn_readlane(c, 2) & 0xFFFFu) |
                 ((u32)__builtin_amdgcn_readlane(c, 3) << 16);
        u32 i2 = ((u32)__builtin_amdgcn_readlane(c, 4) & 0xFFFFu) |
                 ((u32)__builtin_amdgcn_readlane(c, 5) << 16);
        u32 i3 = ((u32)__builtin_amdgcn_readlane(c, 6) & 0xFFFFu) |
                 ((u32)__builtin_amdgcn_readlane(c, 7) << 16);
        int rem = len - ch * RPG;
        if (rem > RPG) rem = RPG;
        rem = __builtin_amdgcn_readfirstlane(rem);

        // D# group 0: count=1, gather_mode=1 (16-bit idx), lds_addr, global_addr, type=2
        v4u g0;
        g0.x = 1u | (1u << 31);
        g0.y = (ch & 1) ? lds1 : lds0;
        g0.z = (u32)gbase;
        g0.w = ((u32)(gbase >> 32) & 0x01FFFFFFu) | (2u << 30);
        // D# group 1: data_size=4B, tensor_dim0=256, tensor_dim1=16384,
        //             tile_dim0=256, tile_dim1=#indices, dim0_stride=256
        v8i g1 = { (int)(2u << 16),
                   (int)(((u32)N_DIM & 0xFFFFu) << 16),
                   (int)(((u32)K_DIM & 0xFFFFu) << 16),
                   (int)((u32)N_DIM << 16),
                   rem,
                   N_DIM,
                   0, 0 };
        v4i g2 = { (int)i0, (int)i1, (int)i2, (int)i3 };  // 16-bit row indices 0..7
        v4i g3 = { 0, 0, 0, 0 };
        v8i gx = { 0, 0, 0, 0, 0, 0, 0, 0 };  // extra arg on clang-23 lane (VADDR4 unused per ISA)
        __builtin_amdgcn_tensor_load_to_lds(g0, g1, g2, g3, gx, 0);
    };

    if (nch > 0) issue(0);
    for (int ch = 0; ch < nch; ++ch) {
        if (ch + 1 < nch) {
            issue(ch + 1);
            __builtin_amdgcn_s_wait_tensorcnt(1);  // chunk ch complete
        } else {
            __builtin_amdgcn_s_wait_tensorcnt(0);
        }
        int rem = len - ch * RPG;
        if (rem > RPG) rem = RPG;
        const float* b = (ch & 1) ? buf1 : buf0;
        for (int k = 0; k < rem; ++k) {
            float v = sval[s + ch * RPG + k];
            const float4* rp = (const float4*)&b[k * N_DIM + lane * 8];
            float4 d0 = rp[0], d1 = rp[1];
            acc0.x = fmaf(v, d0.x, acc0.x); acc0.y = fmaf(v, d0.y, acc0.y);
            acc0.z = fmaf(v, d0.z, acc0.z); acc0.w = fmaf(v, d0.w, acc0.w);
            acc1.x = fmaf(v, d1.x, acc1.x); acc1.y = fmaf(v, d1.y, acc1.y);
            acc1.z = fmaf(v, d1.z, acc1.z); acc1.w = fmaf(v, d1.w, acc1.w);
        }
    }

    const float4* ip = (const float4*)(input + (size_t)row * N_DIM);
    float4 a0 = ip[lane * 2], a1 = ip[lane * 2 + 1];
    float4 o0 = make_float4(a0.x + acc0.x, a0.y + acc0.y, a0.z + acc0.z, a0.w + acc0.w);
    float4 o1 = make_float4(a1.x + acc1.x, a1.y + acc1.y, a1.z + acc1.z, a1.w + acc1.w);
    float4* op = (float4*)(out + (size_t)row * N_DIM);
    op[lane * 2] = o0;
    op[lane * 2 + 1] = o1;
}
#endif  // HAVE_TDM

__global__ void __launch_bounds__(256) spmm_rows_plain(
    const float* __restrict__ input, const float* __restrict__ dense,
    const int* __restrict__ seg, const float* __restrict__ sval,
    const int* __restrict__ scol, float* __restrict__ out) {
    const int lane = threadIdx.x & 31;
    const int wave = threadIdx.x >> 5;
    const int row  = blockIdx.x * 8 + wave;
    const int s = seg[row], e = seg[row + 1];

    float4 acc0 = make_float4(0.f, 0.f, 0.f, 0.f);
    float4 acc1 = make_float4(0.f, 0.f, 0.f, 0.f);
    for (int j = s; j < e; ++j) {
        int c = scol[j];
        float v = sval[j];
        if (j + 1 < e) {
            int cn = scol[j + 1];
            __builtin_prefetch(dense + (size_t)cn * N_DIM + lane * 8, 0, 1);
        }
        const float4* dp = (const float4*)(dense + (size_t)c * N_DIM);
        float4 d0 = dp[lane * 2], d1 = dp[lane * 2 + 1];
        acc0.x = fmaf(v, d0.x, acc0.x); acc0.y = fmaf(v, d0.y, acc0.y);
        acc0.z = fmaf(v, d0.z, acc0.z); acc0.w = fmaf(v, d0.w, acc0.w);
        acc1.x = fmaf(v, d1.x, acc1.x); acc1.y = fmaf(v, d1.y, acc1.y);
        acc1.z = fmaf(v, d1.z, acc1.z); acc1.w = fmaf(v, d1.w, acc1.w);
    }

    const float4* ip = (const float4*)(input + (size_t)row * N_DIM);
    float4 a0 = ip[lane * 2], a1 = ip[lane * 2 + 1];
    float4 o0 = make_float4(a0.x + acc0.x, a0.y + acc0.y, a0.z + acc0.z, a0.w + acc0.w);
    float4 o1 = make_float4(a1.x + acc1.x, a1.y + acc1.y, a1.z + acc1.z, a1.w + acc1.w);
    float4* op = (float4*)(out + (size_t)row * N_DIM);
    op[lane * 2] = o0;
    op[lane * 2 + 1] = o1;
}

// ---------------- fallback (workspace too small): atomic scatter ----------------

__global__ void copy_in(const float* __restrict__ in, float* __restrict__ out) {
    size_t i = (size_t)blockIdx.x * blockDim.x + threadIdx.x;
    out[i] = in[i];
}

__global__ void atomic_spmm(const float* __restrict__ values, const int* __restrict__ rows,
                            const int* __restrict__ cols, const float* __restrict__ dense,
                            float* __restrict__ out) {
    int nz = blockIdx.x * 4 + (threadIdx.x >> 6);  // 4 nnz per 256-thread block
    int t = threadIdx.x & 63;                      // 64 lanes x 4 floats = 256
    float v = values[nz];
    int r = rows[nz], c = cols[nz];
    const float* dp = dense + (size_t)c * N_DIM + t * 4;
    float* op = out + (size_t)r * N_DIM + t * 4;
#pragma unroll
    for (int k = 0; k < 4; ++k) atomicAdd(op + k, v * dp[k]);
}

// ---------------- launcher ----------------

extern "C" void kernel_launch(void* const* d_in, const int* in_sizes, int n_in,
                              void* d_out, int out_size, void* d_ws, size_t ws_size,
                              hipStream_t stream) {
    const float* input  = (const float*)d_in[0];
    const float* values = (const float*)d_in[1];
    const int*   rows   = (const int*)d_in[2];
    const int*   cols   = (const int*)d_in[3];
    const float* dense  = (const float*)d_in[4];
    float* out = (float*)d_out;

    const size_t REQ = 256 * 1024 + (size_t)8 * NNZ_CNT;  // ~8.4 MB
    if (ws_size >= REQ) {
        char* ws = (char*)d_ws;
        int*   counts = (int*)ws;                                   // 16384
        int*   seg    = (int*)(ws + 64 * 1024);                     // 16385
        int*   cursor = (int*)(ws + 160 * 1024);                    // 16384
        float* sval   = (float*)(ws + 256 * 1024);                  // NNZ
        int*   scol   = (int*)(ws + 256 * 1024 + (size_t)4 * NNZ_CNT);

        zero_counts<<<M_DIM / 256, 256, 0, stream>>>(counts);
        count_rows<<<NNZ_CNT / 256, 256, 0, stream>>>(rows, counts);
        scan_counts<<<1, 1024, 0, stream>>>(counts, seg, cursor);
        scatter_nnz<<<NNZ_CNT / 256, 256, 0, stream>>>(values, rows, cols, cursor, sval, scol);
#ifdef HAVE_TDM
        spmm_rows_tdm<<<M_DIM / 8, 256, 8 * 2 * RPG * N_DIM * sizeof(float), stream>>>(
            input, dense, seg, sval, scol, out);
#else
        spmm_rows_plain<<<M_DIM / 8, 256, 0, stream>>>(input, dense, seg, sval, scol, out);
#endif
    } else {
        copy_in<<<(M_DIM * (size_t)N_DIM) / 256, 256, 0, stream>>>(input, out);
        atomic_spmm<<<NNZ_CNT / 4, 256, 0, stream>>>(values, rows, cols, dense, out);
    }
}